// TensorDiffEq_31215822307402
// MI455X (gfx1250) — compile-verified
//
#include <hip/hip_runtime.h>

typedef __attribute__((ext_vector_type(2))) float v2f;
typedef __attribute__((ext_vector_type(8))) float v8f;

// ---------------------------------------------------------------------------
// Kernel 1: W = P_mat(64x64) @ expm(U)(64x64) @ F_pad(64x16), single workgroup.
// expm via scaling-and-squaring: T = U/2^8, E = sum_{k<=12} T^k/k!, E = E^(2^8).
// Output W stored row-major [k][n] (n padded 10->16 with zeros) into d_ws.
// ---------------------------------------------------------------------------
__global__ void __launch_bounds__(256)
tde_prep_kernel(const float* __restrict__ P, const float* __restrict__ U,
                const float* __restrict__ F, float* __restrict__ W) {
    __shared__ float sT[64 * 64];     // scaled U
    __shared__ float sE[64 * 64];     // exp accumulator
    __shared__ float sTerm[64 * 64];  // current Taylor term / scratch
    const int tid = threadIdx.x;
    const float scale = 1.0f / 256.0f;  // 2^-8

    for (int idx = tid; idx < 4096; idx += 256) {
        float t = U[idx] * scale;
        sT[idx] = t;
        int r = idx >> 6, c = idx & 63;
        sE[idx] = t + ((r == c) ? 1.0f : 0.0f);
        sTerm[idx] = t;
    }
    __syncthreads();

    for (int k = 2; k <= 12; ++k) {   // Taylor terms
        const float invk = 1.0f / (float)k;
        float nt[16];
#pragma unroll
        for (int e = 0; e < 16; ++e) {
            int idx = tid * 16 + e;
            int row = idx >> 6, col = idx & 63;
            float acc = 0.0f;
            for (int kk = 0; kk < 64; ++kk)
                acc += sTerm[row * 64 + kk] * sT[kk * 64 + col];
            nt[e] = acc * invk;
        }
        __syncthreads();
#pragma unroll
        for (int e = 0; e < 16; ++e) {
            int idx = tid * 16 + e;
            sTerm[idx] = nt[e];
            sE[idx] += nt[e];
        }
        __syncthreads();
    }

    for (int s = 0; s < 8; ++s) {     // 8 squarings: E = E @ E
        float nt[16];
#pragma unroll
        for (int e = 0; e < 16; ++e) {
            int idx = tid * 16 + e;
            int row = idx >> 6, col = idx & 63;
            float acc = 0.0f;
            for (int kk = 0; kk < 64; ++kk)
                acc += sE[row * 64 + kk] * sE[kk * 64 + col];
            nt[e] = acc;
        }
        __syncthreads();
#pragma unroll
        for (int e = 0; e < 16; ++e) sE[tid * 16 + e] = nt[e];
        __syncthreads();
    }

    {   // W1 = P_mat @ E -> sTerm
        float nt[16];
#pragma unroll
        for (int e = 0; e < 16; ++e) {
            int idx = tid * 16 + e;
            int row = idx >> 6, col = idx & 63;
            float acc = 0.0f;
            for (int kk = 0; kk < 64; ++kk)
                acc += P[row * 64 + kk] * sE[kk * 64 + col];
            nt[e] = acc;
        }
        __syncthreads();
#pragma unroll
        for (int e = 0; e < 16; ++e) sTerm[tid * 16 + e] = nt[e];
        __syncthreads();
    }

    for (int e = 0; e < 4; ++e) {     // W = W1 @ F (pad cols 10..15 with 0)
        int idx = tid * 4 + e;
        int i = idx >> 4, n = idx & 15;
        float acc = 0.0f;
        if (n < 10)
            for (int kk = 0; kk < 64; ++kk)
                acc += sTerm[i * 64 + kk] * F[kk * 10 + n];
        W[idx] = acc;
    }
}

// ---------------------------------------------------------------------------
// Kernel 2: out(B x 10) = x(B x 64) @ W(64 x 16-padded) with
// V_WMMA_F32_16X16X4_F32. Each wave: one 16x64 tile of x per step, staged
// through LDS by async DMA (global_load_async_to_lds_b128, ASYNCcnt), double
// buffered. LDS rows padded to 68 floats -> conflict-free ds_load_b64
// fragment reads. Global side is fully coalesced (each 128B line once).
// ---------------------------------------------------------------------------
#define TDE_WAVES      8
#define TDE_ROWSTRIDE  68                      // floats per LDS row (padded)
#define TDE_BUF_FLTS   (16 * TDE_ROWSTRIDE)    // 1088 floats = 4352 B
#define TDE_WAVE_FLTS  (2 * TDE_BUF_FLTS)      // double buffer
#define TDE_LDS_BYTES  (TDE_WAVES * TDE_WAVE_FLTS * 4)  // 69632 B

__device__ __forceinline__ void tde_async_tile(const float* x, int tile,
                                               unsigned ldsBufBase, int lane) {
    // lane l fetches 16 bytes at tile_base + l*16 + j*512 (fully coalesced);
    // that 16B chunk is row r = 2j + (l>>4), cols 4*(l&15)..+3 of the tile.
    unsigned long long ga = (unsigned long long)(uintptr_t)(x) +
                            (unsigned long long)tile * 4096ull +
                            (unsigned)(lane * 16);
    unsigned ldsa = ldsBufBase + (unsigned)((lane >> 4) * (TDE_ROWSTRIDE * 4) +
                                            (lane & 15) * 16);
#pragma unroll
    for (int j = 0; j < 8; ++j) {
        asm volatile("global_load_async_to_lds_b128 %0, %1, off"
                     :: "v"(ldsa + (unsigned)(2 * TDE_ROWSTRIDE * 4) * j),
                        "v"(ga + 512ull * j)
                     : "memory");
    }
}

__global__ void __launch_bounds__(256)
tde_fold_gemm_kernel(const float* __restrict__ x, const float* __restrict__ W,
                     float* __restrict__ out, int numTiles) {
    extern __shared__ float smem[];
    const int lane  = threadIdx.x & 31;
    const int waveI = threadIdx.x >> 5;
    const int n     = lane & 15;        // B/C/D lane column
    const int khalf = (lane >> 4) * 2;  // lanes 0-15 -> K{0,1}; 16-31 -> K{2,3}
    const int m     = lane & 15;        // A-matrix row for this lane

    // B-fragments for W (64x16 row-major), resident in VGPRs for all tiles.
    v2f bfrag[16];
#pragma unroll
    for (int kk = 0; kk < 16; ++kk) {
        int k0 = kk * 4 + khalf;
        bfrag[kk].x = W[(k0 + 0) * 16 + n];
        bfrag[kk].y = W[(k0 + 1) * 16 + n];
    }

    float* wavebuf = smem + waveI * TDE_WAVE_FLTS;
    const unsigned ldsWaveBase = (unsigned)(unsigned long long)(uintptr_t)wavebuf;

    const int stride = gridDim.x * TDE_WAVES;
    int tile = blockIdx.x * TDE_WAVES + waveI;
    int buf = 0;
    if (tile < numTiles) tde_async_tile(x, tile, ldsWaveBase, lane);

    for (; tile < numTiles; tile += stride) {
        const int next = tile + stride;
        if (next < numTiles) {
            tde_async_tile(x, next, ldsWaveBase + (unsigned)((buf ^ 1) * TDE_BUF_FLTS * 4), lane);
            // 8 new outstanding; in-order completion => current tile resident.
            asm volatile("s_wait_asynccnt 0x8" ::: "memory");
        } else {
            asm volatile("s_wait_asynccnt 0x0" ::: "memory");
        }

        const float* sbuf = wavebuf + buf * TDE_BUF_FLTS;
        v8f c = {};
#pragma unroll
        for (int kk = 0; kk < 16; ++kk) {
            // Conflict-free ds_load_b64: banks 4*(m+kk)+khalf all distinct.
            v2f a = *(const v2f*)(sbuf + m * TDE_ROWSTRIDE + kk * 4 + khalf);
            c = __builtin_amdgcn_wmma_f32_16x16x4_f32(
                    false, a, false, bfrag[kk], (short)0, c, false, false);
        }

        // C/D layout: lanes 0-15 -> M = r, lanes 16-31 -> M = r + 8 (VGPR r).
        const int mbase = (lane >> 4) * 8;
        if (n < 10) {
            float* o = out + ((size_t)tile * 16 + mbase) * 10 + n;
#pragma unroll
            for (int r = 0; r < 8; ++r) o[(size_t)r * 10] = c[r];
        }
        buf ^= 1;
    }
}

extern "C" void kernel_launch(void* const* d_in, const int* in_sizes, int n_in,
                              void* d_out, int out_size, void* d_ws, size_t ws_size,
                              hipStream_t stream) {
    const float* x = (const float*)d_in[0];  // (B, 64)
    const float* P = (const float*)d_in[1];  // (64, 4,4,4)
    const float* U = (const float*)d_in[2];  // (4,4,4, 4,4,4)
    const float* F = (const float*)d_in[3];  // (4,4,4, 10)
    float* out = (float*)d_out;              // (B, 10)
    float* W   = (float*)d_ws;               // 64*16 floats = 4 KB scratch

    const int B = in_sizes[0] / 64;          // 524288
    const int numTiles = B / 16;             // 32768

    tde_prep_kernel<<<1, 256, 0, stream>>>(P, U, F, W);
    const int blocks = 1024;                 // 8192 waves, 4 tiles each (pipelined)
    tde_fold_gemm_kernel<<<blocks, 256, TDE_LDS_BYTES, stream>>>(x, W, out, numTiles);
}